// Model_51453708206351
// MI455X (gfx1250) — compile-verified
//
#include <hip/hip_runtime.h>
#include <stdint.h>

// Problem constants (from reference): T=32768, H2=4096, H=2048, G=8
#define TT   32768
#define HH   2048
#define HH2  4096
#define GG   8
#define ROWS_PER_BLOCK 16
#define THREADS 256

typedef float        f32x4 __attribute__((ext_vector_type(4)));
typedef unsigned int u32x4 __attribute__((ext_vector_type(4)));
typedef int          i32x4 __attribute__((ext_vector_type(4)));
typedef int          i32x8 __attribute__((ext_vector_type(8)));

#if defined(__AMDGCN__) && __has_builtin(__builtin_amdgcn_tensor_load_to_lds) && \
    __has_builtin(__builtin_amdgcn_s_wait_tensorcnt)
#define HAVE_TDM 1
#else
#define HAVE_TDM 0
#endif

#if HAVE_TDM
// Issue one TDM row load: 4096 fp32 (16 KB) from global -> LDS.
// D# layout per CDNA5 ISA ch.8 (CDNA5 ISA).
__device__ __forceinline__ void tdm_load_row(const float* gsrc, unsigned lds_byte) {
  unsigned long long ga = (unsigned long long)(uintptr_t)gsrc;
  u32x4 g0;
  g0.x = 1u;                                            // count=1 (valid descriptor)
  g0.y = lds_byte;                                      // lds_addr [63:32]
  g0.z = (unsigned)(ga & 0xffffffffull);                // global_addr[31:0]  (bits 95:64)
  g0.w = (unsigned)((ga >> 32) & 0x01ffffffull)         // global_addr[56:32] (bits 120:96)
       | (2u << 30);                                    // type=2 ("image")
  i32x8 g1;
  g1[0] = (2 << 16);                                    // workgroup_mask=0, data_size=2 (4B)
  g1[1] = (int)((HH2 & 0xffffu) << 16);                 // tensor_dim0[15:0] @ [31:16]
  g1[2] = (int)((HH2 >> 16) & 0xffffu) | (1 << 16);     // tensor_dim0[31:16], tensor_dim1=1
  g1[3] = (int)(((unsigned)HH2) << 16);                 // tile_dim0 = 4096 @ [31:16]
  g1[4] = 0;                                            // tile_dim1=0 (unused), tile_dim2=0
  g1[5] = (int)HH2;                                     // tensor_dim0_stride[31:0]
  g1[6] = 0;                                            // stride hi, dim1_stride lo
  g1[7] = 0;
  i32x4 z4 = {0, 0, 0, 0};
#if __has_include(<hip/amd_detail/amd_gfx1250_TDM.h>)
  // therock / clang-23 toolchain: 6-arg form
  i32x8 z8 = {0, 0, 0, 0, 0, 0, 0, 0};
  __builtin_amdgcn_tensor_load_to_lds(g0, g1, z4, z4, z8, 0);
#else
  // ROCm 7.2 / clang-22 toolchain: 5-arg form
  __builtin_amdgcn_tensor_load_to_lds(g0, g1, z4, z4, 0);
#endif
}
#endif // HAVE_TDM

__device__ __forceinline__ float silu_mul(float l, float r) {
  // l * (r * sigmoid(r))
  float sig = 1.0f / (1.0f + __expf(-r));
  return l * (r * sig);
}

__global__ __launch_bounds__(THREADS) void swiglu_smoothquant(
    const float* __restrict__ x,        // (T, H2)
    const float* __restrict__ smooth,   // (G, H)
    const int*   __restrict__ gindex,   // (G,)
    const int*   __restrict__ glt,      // group_list_type scalar
    float*       __restrict__ qout,     // (T, H) quantized values (stored as float)
    float*       __restrict__ dyn_out)  // (T,)
{
  const int tid  = threadIdx.x;
  const int row0 = blockIdx.x * ROWS_PER_BLOCK;

#if HAVE_TDM
  __shared__ f32x4 buf[2][HH2 / 4];     // 2 x 16 KB double buffer
#endif
  __shared__ float red[9];

  // Group ends (uniform across block; group_list_type==1 -> cumsum)
  int ends[GG];
  {
    const bool cum = (glt[0] == 1);
    int acc = 0;
#pragma unroll
    for (int k = 0; k < GG; ++k) {
      int v = gindex[k];
      acc += v;
      ends[k] = cum ? acc : v;
    }
  }

#if HAVE_TDM
  if (tid < 32) {  // wave 0 drives the TDM pipeline
    tdm_load_row(x + (size_t)row0 * HH2, (unsigned)(uintptr_t)&buf[0][0]);
  }
#endif

  for (int i = 0; i < ROWS_PER_BLOCK; ++i) {
    const int row = row0 + i;

#if HAVE_TDM
    if (tid < 32) {
      if (i + 1 < ROWS_PER_BLOCK) {
        tdm_load_row(x + (size_t)(row + 1) * HH2,
                     (unsigned)(uintptr_t)&buf[(i + 1) & 1][0]);
        __builtin_amdgcn_s_wait_tensorcnt(1);  // oldest (current row) done; prefetch in flight
      } else {
        __builtin_amdgcn_s_wait_tensorcnt(0);
      }
    }
    __syncthreads();  // publish TDM-written LDS to all 8 waves
    {
      unsigned bp = (unsigned)(uintptr_t)&buf[0][0];
      asm volatile("" ::"v"(bp) : "memory");  // TDM writes LDS behind compiler's back
    }
    const f32x4* b = &buf[i & 1][0];
    f32x4 l0 = b[tid];
    f32x4 l1 = b[tid + 256];
    f32x4 r0 = b[512 + tid];
    f32x4 r1 = b[512 + tid + 256];
#else
    const f32x4* xr = (const f32x4*)(x + (size_t)row * HH2);
    f32x4 l0 = __builtin_nontemporal_load(xr + tid);
    f32x4 l1 = __builtin_nontemporal_load(xr + tid + 256);
    f32x4 r0 = __builtin_nontemporal_load(xr + 512 + tid);
    f32x4 r1 = __builtin_nontemporal_load(xr + 512 + tid + 256);
#endif

    // gid = searchsorted(ends, row, side='right') = #(ends[k] <= row)
    int gid = 0;
#pragma unroll
    for (int k = 0; k < GG; ++k) gid += (ends[k] <= row) ? 1 : 0;

    f32x4 s0 = {1.f, 1.f, 1.f, 1.f}, s1 = {1.f, 1.f, 1.f, 1.f};
    if (gid < GG) {  // block-uniform branch; scales table (64 KB) stays L2/WGP$ resident
      const f32x4* sp = (const f32x4*)(smooth + (size_t)gid * HH);
      s0 = sp[tid];
      s1 = sp[tid + 256];
    }

    f32x4 o0, o1;
#pragma unroll
    for (int c = 0; c < 4; ++c) {
      o0[c] = silu_mul(l0[c], r0[c]) * s0[c];
      o1[c] = silu_mul(l1[c], r1[c]) * s1[c];
    }

    // Row abs-max: per-thread -> wave32 shuffle tree -> cross-wave via LDS
    float m = 0.0f;
#pragma unroll
    for (int c = 0; c < 4; ++c)
      m = fmaxf(m, fmaxf(fabsf(o0[c]), fabsf(o1[c])));
#pragma unroll
    for (int off = 16; off > 0; off >>= 1)
      m = fmaxf(m, __shfl_xor(m, off, 32));
    if ((tid & 31) == 0) red[tid >> 5] = m;
    __syncthreads();
    if (tid == 0) {
      float mm = red[0];
#pragma unroll
      for (int w = 1; w < 8; ++w) mm = fmaxf(mm, red[w]);
      red[8] = 127.0f / fmaxf(mm, 1e-10f);
    }
    __syncthreads();
    const float dyn = red[8];
    if (tid == 0) dyn_out[row] = dyn;

    // clip then round-to-nearest-even (rintf, default RTE == jnp.round)
    f32x4 q0, q1;
#pragma unroll
    for (int c = 0; c < 4; ++c) {
      q0[c] = rintf(fminf(fmaxf(o0[c] * dyn, -128.0f), 127.0f));
      q1[c] = rintf(fminf(fmaxf(o1[c] * dyn, -128.0f), 127.0f));
    }
    f32x4* qr = (f32x4*)(qout + (size_t)row * HH);
    __builtin_nontemporal_store(q0, qr + tid);        // streamed output: bypass/NT cache hint
    __builtin_nontemporal_store(q1, qr + tid + 256);
  }
}

extern "C" void kernel_launch(void* const* d_in, const int* in_sizes, int n_in,
                              void* d_out, int out_size, void* d_ws, size_t ws_size,
                              hipStream_t stream) {
  (void)in_sizes; (void)n_in; (void)out_size; (void)d_ws; (void)ws_size;
  // setup_inputs order: x, smooth_scales, group_index, quant_mode, group_list_type
  const float* x      = (const float*)d_in[0];
  const float* smooth = (const float*)d_in[1];
  const int*   gidx   = (const int*)d_in[2];
  const int*   glt    = (const int*)d_in[4];  // d_in[3] = quant_mode (unused by reference math)

  float* qout = (float*)d_out;                 // T*H quantized values (output dtype rule: float)
  float* dyn  = qout + (size_t)TT * HH;        // then T dyn_scale values

  dim3 grid(TT / ROWS_PER_BLOCK);
  dim3 block(THREADS);
  swiglu_smoothquant<<<grid, block, 0, stream>>>(x, smooth, gidx, glt, qout, dyn);
}